// GauntTwoBody_68642167324708
// MI455X (gfx1250) — compile-verified
//
#include <hip/hip_runtime.h>

#define LMAXQ   4
#define CIN     64
#define COUTN   64
#define GRID9   9
#define NI      25
#define CA      128            // CIN * VMAX
#define UV      81             // 9*9
#define N1W     (5*64*128)     // 40960 w1 floats / sample
#define NCW     (5*64*64)      // 20480 wc floats / sample
#define WSTRIDE (N1W + NCW)    // 61440
#define FINN    (CIN*NI)       // 1600
#define FOUTN   (COUTN*NI)     // 1600
#define STR     68             // padded c-stride: banks (4n+2kh)%64 all distinct

typedef __attribute__((ext_vector_type(2))) float v2f;
typedef __attribute__((ext_vector_type(8))) float v8f;

// ---- dynamic LDS layout (in floats) -------------------------------------
#define N_PADT      (5*16*STR)               // 5440  padded transposed block
#define OFF_FEATT   0                        // featT  [5][16][STR]
#define OFF_FEAT1   (N_PADT)                 // feat1  [128][25] = 3200
#define OFF_FEAT2T  (N_PADT + 3200)          // feat2T [5][16][STR]
#define OFF_CPLX    (2*N_PADT + 3200)        // float2 region (even -> 8B aligned)
#define N_CUV       2025
#define N_IM        (CA*UV)                  // 10368
#define N_B2        (64*UV)                  // 5184
#define SMEM_FLOATS (OFF_CPLX + 2*(N_CUV + N_CUV + N_IM + N_B2))
#define SMEM_BYTES  (SMEM_FLOATS * 4)        // 213,136 B < 320 KB

__global__ __launch_bounds__(256, 1)
void gaunt_two_body_kernel(const float* __restrict__ feature,
                           const float* __restrict__ weight,
                           const float* __restrict__ Cg_,   // complex64 interleaved
                           const float* __restrict__ Rg_,
                           float* __restrict__ out)
{
    extern __shared__ float smem[];
    float*  featT  = smem + OFF_FEATT;
    float*  feat1  = smem + OFF_FEAT1;
    float*  feat2T = smem + OFF_FEAT2T;
    float2* Cuv = (float2*)(smem + OFF_CPLX);
    float2* Ruv = Cuv + N_CUV;
    float2* im  = Ruv + N_CUV;
    float2* B2s = im  + N_IM;

    const int b    = blockIdx.x;
    const int tid  = threadIdx.x;
    const int lane = tid & 31;
    const int wave = tid >> 5;
    const int mn   = lane & 15;   // M for A-frag, N for B-frag / C-frag
    const int kh   = lane >> 4;   // which K-pair half (0/1)

    const float* w1  = weight + (size_t)b * WSTRIDE;   // [5][64][128]
    const float* wc  = w1 + N1W;                       // [5][64][64]
    const float* fin = feature + (size_t)b * FINN;

    // ---- Phase 0a: zero padded transposed buffers, stage Gaunt tensors ----
    for (int i = tid; i < N_PADT; i += 256) { featT[i] = 0.f; feat2T[i] = 0.f; }
    const float2* Cg = (const float2*)Cg_;
    const float2* Rg = (const float2*)Rg_;
    for (int i = tid; i < N_CUV; i += 256) { Cuv[i] = Cg[i]; Ruv[i] = Rg[i]; }
    __syncthreads();

    // ---- Phase 0b: stage feature zero-padded + K-transposed:
    //      featT[(l*16 + n)*STR + c] = f_l[c][n]   (n >= d stays 0)
    #define STAGE_L(l_, d_)                                                 \
    {                                                                       \
        const int off = CIN*(l_)*(l_);                                      \
        for (int j = tid; j < CIN*(d_); j += 256) {                         \
            const int c = j / (d_);                                         \
            const int n = j - c*(d_);                                       \
            featT[((l_)*16 + n)*STR + c] = fin[off + j];                    \
        }                                                                   \
    }
    STAGE_L(0,1) STAGE_L(1,3) STAGE_L(2,5) STAGE_L(3,7) STAGE_L(4,9)
    #undef STAGE_L
    __syncthreads();

    // ---- Phase 1: feat1[a][l^2+n] = sum_c w1[l][c][a] * f_l[c][n]  (WMMA) --
    // wave w owns M-tile rows a in [16w, 16w+16); K = 64 in 16 steps of 4.
    {
        for (int l = 0; l <= LMAXQ; ++l) {
            const int d = 2*l + 1;
            const float* wl  = w1 + l * (64*CA);
            const float* ftl = featT + (l*16 + mn)*STR;   // per-lane K row pair base
            v8f acc0 = {0.f,0.f,0.f,0.f,0.f,0.f,0.f,0.f};
            v8f acc1 = {0.f,0.f,0.f,0.f,0.f,0.f,0.f,0.f};
            #pragma unroll
            for (int k0 = 0; k0 < 64; k0 += 8) {
                {   // chain 0: K = k0 .. k0+3
                    const int ka = k0 + 2*kh;
                    v2f a;
                    a.x = wl[(ka+0)*CA + wave*16 + mn];
                    a.y = wl[(ka+1)*CA + wave*16 + mn];
                    const v2f bf = *(const v2f*)(ftl + ka);   // ds_load_b64
                    acc0 = __builtin_amdgcn_wmma_f32_16x16x4_f32(
                        false, a, false, bf, (short)0, acc0, false, false);
                }
                {   // chain 1: K = k0+4 .. k0+7
                    const int ka = k0 + 4 + 2*kh;
                    v2f a;
                    a.x = wl[(ka+0)*CA + wave*16 + mn];
                    a.y = wl[(ka+1)*CA + wave*16 + mn];
                    const v2f bf = *(const v2f*)(ftl + ka);
                    acc1 = __builtin_amdgcn_wmma_f32_16x16x4_f32(
                        false, a, false, bf, (short)0, acc1, false, false);
                }
            }
            if (mn < d) {
                #pragma unroll
                for (int r = 0; r < 8; ++r) {
                    const int gm = wave*16 + r + 8*kh;     // C/D: M = r + 8*(lane/16)
                    feat1[gm*NI + l*l + mn] = acc0[r] + acc1[r];
                }
            }
        }
    }
    __syncthreads();

    // ---- Phase 2: project to uv grid: im[a][uv] = sum_i feat1[a][i]*Ciuv[i][uv]
    for (int idx = tid; idx < CA*UV; idx += 256) {
        const int a  = idx / UV;
        const int uv = idx - a*UV;
        const float* fr = feat1 + a*NI;
        float re = 0.f, imm = 0.f;
        #pragma unroll
        for (int i = 0; i < NI; ++i) {
            const float  f  = fr[i];
            const float2 cc = Cuv[i*UV + uv];
            re  = fmaf(f, cc.x, re);
            imm = fmaf(f, cc.y, imm);
        }
        im[idx] = make_float2(re, imm);
    }
    __syncthreads();

    // ---- Phase 3: FFT-free Gaunt product = full 2-D linear convolution,
    //      cropped to [0:9,0:9] (17 = 2*9-1 so circular == linear).
    for (int idx = tid; idx < 64*UV; idx += 256) {
        const int c  = idx / UV;
        const int uv = idx - c*UV;
        const int u  = uv / GRID9;
        const int v  = uv - u*GRID9;
        const float2* p0 = im + c*UV;            // v-component 0
        const float2* p1 = im + (64 + c)*UV;     // v-component 1
        float re = 0.f, ii = 0.f;
        for (int du = 0; du <= u; ++du) {
            const float2* r0 = p0 + du*GRID9;
            const float2* r1 = p1 + (u - du)*GRID9;
            for (int dv = 0; dv <= v; ++dv) {
                const float2 x = r0[dv];
                const float2 y = r1[v - dv];
                re = fmaf(x.x, y.x, fmaf(-x.y, y.y, re));
                ii = fmaf(x.x, y.y, fmaf( x.y, y.x, ii));
            }
        }
        B2s[idx] = make_float2(re, ii);
    }
    __syncthreads();

    // ---- Phase 4: back-project (real part) straight into padded transposed
    //      layout: feat2T[(l*16+n)*STR + c] = sum_uv Re(B2[c]*R[i]),  i = l^2+n
    for (int idx = tid; idx < 64*NI; idx += 256) {
        const int c = idx / NI;
        const int i = idx - c*NI;
        const int l = (i>=1) + (i>=4) + (i>=9) + (i>=16);
        const int n = i - l*l;
        const float2* bp = B2s + c*UV;
        const float2* rp = Ruv + i*UV;
        float acc = 0.f;
        #pragma unroll 3
        for (int uv = 0; uv < UV; ++uv) {
            const float2 x = bp[uv];
            const float2 y = rp[uv];
            acc = fmaf(x.x, y.x, fmaf(-x.y, y.y, acc));
        }
        feat2T[(l*16 + n)*STR + c] = acc;
    }
    __syncthreads();

    // ---- Phase 5: out_l[a][n] = sum_c wc[l][c][a] * feat2[c][l^2+n]  (WMMA)
    // 20 (l, mtile) jobs over 8 waves; waves past 20 redo a job (same data).
    {
        #pragma unroll
        for (int jj = 0; jj < 3; ++jj) {
            int job = wave + 8*jj;
            if (job >= 20) job -= 8;             // duplicate, identical output
            const int l  = job >> 2;
            const int mt = job & 3;
            const int d  = 2*l + 1;
            const float* wl  = wc + l * (64*COUTN);
            const float* f2l = feat2T + (l*16 + mn)*STR;
            v8f acc0 = {0.f,0.f,0.f,0.f,0.f,0.f,0.f,0.f};
            v8f acc1 = {0.f,0.f,0.f,0.f,0.f,0.f,0.f,0.f};
            #pragma unroll
            for (int k0 = 0; k0 < 64; k0 += 8) {
                {
                    const int ka = k0 + 2*kh;
                    v2f a;
                    a.x = wl[(ka+0)*COUTN + mt*16 + mn];
                    a.y = wl[(ka+1)*COUTN + mt*16 + mn];
                    const v2f bf = *(const v2f*)(f2l + ka);
                    acc0 = __builtin_amdgcn_wmma_f32_16x16x4_f32(
                        false, a, false, bf, (short)0, acc0, false, false);
                }
                {
                    const int ka = k0 + 4 + 2*kh;
                    v2f a;
                    a.x = wl[(ka+0)*COUTN + mt*16 + mn];
                    a.y = wl[(ka+1)*COUTN + mt*16 + mn];
                    const v2f bf = *(const v2f*)(f2l + ka);
                    acc1 = __builtin_amdgcn_wmma_f32_16x16x4_f32(
                        false, a, false, bf, (short)0, acc1, false, false);
                }
            }
            if (mn < d) {
                float* op = out + (size_t)b*FOUTN + COUTN*(l*l);
                #pragma unroll
                for (int r = 0; r < 8; ++r) {
                    const int aout = mt*16 + r + 8*kh;
                    op[aout*d + mn] = acc0[r] + acc1[r];
                }
            }
        }
    }
}

extern "C" void kernel_launch(void* const* d_in, const int* in_sizes, int n_in,
                              void* d_out, int out_size, void* d_ws, size_t ws_size,
                              hipStream_t stream) {
    (void)n_in; (void)d_ws; (void)ws_size; (void)out_size;
    const float* feature = (const float*)d_in[0];
    const float* weight  = (const float*)d_in[1];
    const float* Ciuv    = (const float*)d_in[2];   // complex64 interleaved
    const float* RCuvi   = (const float*)d_in[3];
    float* out = (float*)d_out;

    const int B = in_sizes[0] / FINN;               // 2048

    static_assert(SMEM_BYTES <= 320*1024, "LDS budget");
    (void)hipFuncSetAttribute((const void*)gaunt_two_body_kernel,
                              hipFuncAttributeMaxDynamicSharedMemorySize,
                              SMEM_BYTES);

    gaunt_two_body_kernel<<<B, 256, SMEM_BYTES, stream>>>(
        feature, weight, Ciuv, RCuvi, out);
}